// CRF_87866440942111
// MI455X (gfx1250) — compile-verified
//
#include <hip/hip_runtime.h>

typedef __attribute__((ext_vector_type(2))) float v2f;
typedef __attribute__((ext_vector_type(8))) float v8f;

#define N_ 64
#define S_ 1024
#define D_ 1024
#define C_ 128

// ---------------------------------------------------------------------------
// Kernel 0: transpose T (C x C) into ws so backtrace reads T^T[pc] coalesced.
// ---------------------------------------------------------------------------
__global__ void crf_transpose(const float* __restrict__ Tg, float* __restrict__ Tt) {
    // Tt[p*C + c] = Tg[c*C + p]
    Tt[threadIdx.x * C_ + blockIdx.x] = Tg[blockIdx.x * C_ + threadIdx.x];
}

// ---------------------------------------------------------------------------
// Kernel 1: npot0 = X (65536 x 1024) @ W (1024 x 128), f32 WMMA 16x16x4.
// One wave per 16-row M tile, 8 N-tiles of 16 accumulated per wave.
// A layout (16x4 f32): lane=16h+m -> a[j] = X[row0+m][k0+2h+j]
// B layout (4x16 f32): lane=16h+m -> b[j] = W[k0+2h+j][col0+m]
// C/D layout:          lane=16h+m, vgpr r -> D[r+8h][m]
// ---------------------------------------------------------------------------
__global__ __launch_bounds__(256) void crf_gemm(const float* __restrict__ X,
                                                const float* __restrict__ W,
                                                float* __restrict__ npot) {
    const int wv   = threadIdx.x >> 5;
    const int lane = threadIdx.x & 31;
    const int h    = lane >> 4;
    const int m    = lane & 15;
    const int tile = blockIdx.x * 8 + wv;            // 0..4095 (M tiles)

    const float* xrow = X + (size_t)(tile * 16 + m) * D_ + 2 * h;

    v8f acc[8] = {};
    for (int k0 = 0; k0 < D_; k0 += 4) {
        v2f av = *(const v2f*)(xrow + k0);
        const float* wp = W + (k0 + 2 * h) * C_ + m;
#pragma unroll
        for (int nt = 0; nt < 8; ++nt) {
            v2f bv;
            bv.x = wp[nt * 16];
            bv.y = wp[nt * 16 + C_];
            acc[nt] = __builtin_amdgcn_wmma_f32_16x16x4_f32(
                false, av, false, bv, (short)0, acc[nt], false, false);
        }
    }
#pragma unroll
    for (int nt = 0; nt < 8; ++nt) {
#pragma unroll
        for (int r = 0; r < 8; ++r) {
            npot[(size_t)(tile * 16 + r + 8 * h) * C_ + nt * 16 + m] = acc[nt][r];
        }
    }
}

// ---------------------------------------------------------------------------
// Kernel 2: fused forward-fill + Viterbi forward scan. One block per batch n.
// 256 threads = two halves; half 'half' reduces max over c in [half*64,half*64+64)
// with its T rows held in registers. Lower half owns state l[c], cur_np[c].
// Writes: filled npot (in-place), l[t] rows, score, last path entry, argmax seed.
// ---------------------------------------------------------------------------
__global__ __launch_bounds__(256) void crf_forward(float* __restrict__ npot,
                                                   const float* __restrict__ mask,
                                                   const float* __restrict__ Tg,
                                                   float* __restrict__ lbuf,
                                                   float* __restrict__ pred,
                                                   float* __restrict__ score,
                                                   int* __restrict__ prevc) {
    __shared__ __align__(16) float vlds[C_];
    __shared__ float plds[2 * C_];
    __shared__ float mlds[S_];

    const int tid  = threadIdx.x;
    const int half = tid >> 7;
    const int c    = tid & (C_ - 1);
    const int n    = blockIdx.x;
    const int cbase = half * 64;

    for (int t = tid; t < S_; t += 256) mlds[t] = mask[n * S_ + t];

    float treg[64];
#pragma unroll
    for (int j = 0; j < 64; ++j) treg[j] = Tg[(cbase + j) * C_ + c];

    float l = 0.0f;
    float cur = 0.0f;
    if (half == 0) cur = npot[((size_t)n * S_ + (S_ - 1)) * C_ + c];  // ff init = npot0[:, -1]
    __syncthreads();

    for (int t = 0; t < S_ - 1; ++t) {
        if (half == 0) {
            float mt   = mlds[t];
            float orig = npot[((size_t)n * S_ + t) * C_ + c];
            cur = (mt != 0.0f) ? orig : cur;                       // forward fill
            npot[((size_t)n * S_ + t) * C_ + c] = cur;             // store filled npot
            lbuf[((size_t)t * N_ + n) * C_ + c] = l;               // store l[t] (pre-step)
            vlds[c] = cur + l;                                     // v = npot[t] + l_prev
        }
        __syncthreads();

        float best = -3.402823466e38f;
#pragma unroll
        for (int j = 0; j < 64; j += 4) {
            float4 vv = *(const float4*)&vlds[cbase + j];
            float s0 = vv.x + treg[j + 0];
            float s1 = vv.y + treg[j + 1];
            float s2 = vv.z + treg[j + 2];
            float s3 = vv.w + treg[j + 3];
            best = fmaxf(best, fmaxf(fmaxf(s0, s1), fmaxf(s2, s3)));
        }
        plds[half * C_ + c] = best;
        __syncthreads();

        if (half == 0) {
            float nw  = fmaxf(plds[c], plds[C_ + c]);
            float mt1 = mlds[t + 1];
            l = (mt1 != 0.0f) ? nw : l;                            // masked update
        }
    }

    // Final timestep: fill npot[S-1], final = npot[S-1] + l[S-1], score/argmax.
    if (half == 0) {
        float mt   = mlds[S_ - 1];
        float orig = npot[((size_t)n * S_ + (S_ - 1)) * C_ + c];
        cur = (mt != 0.0f) ? orig : cur;
        npot[((size_t)n * S_ + (S_ - 1)) * C_ + c] = cur;
        vlds[c] = cur + l;
    }
    __syncthreads();
    if (tid == 0) {
        float best = vlds[0];
        int bi = 0;
        for (int i = 1; i < C_; ++i) {                              // first-index argmax
            float v = vlds[i];
            if (v > best) { best = v; bi = i; }
        }
        score[n] = best;
        float mlast = mlds[S_ - 1];
        pred[n * S_ + (S_ - 1)] = (mlast != 0.0f) ? (float)bi : -1.0f;
        prevc[n] = bi;
    }
}

// ---------------------------------------------------------------------------
// Kernel 3: backtrace. One block (128 threads) per batch. Path-independent
// loads (npot[t], l[t]) are prefetched one step ahead; only T^T[pc] depends
// on the carried state. Argmax with jnp.argmax first-index tie-break.
// ---------------------------------------------------------------------------
__global__ __launch_bounds__(128) void crf_backtrace(const float* __restrict__ npot,
                                                     const float* __restrict__ lbuf,
                                                     const float* __restrict__ Tt,
                                                     const float* __restrict__ mask,
                                                     const int* __restrict__ prevc,
                                                     float* __restrict__ pred) {
    __shared__ float mlds[S_];
    __shared__ float sred[C_];
    __shared__ int pcsh;

    const int tid = threadIdx.x;  // = state c
    const int n   = blockIdx.x;

    for (int t = tid; t < S_; t += 128) mlds[t] = mask[n * S_ + t];
    int pc = prevc[n];

    float np_nx = npot[((size_t)n * S_ + (S_ - 2)) * C_ + tid];
    float l_nx  = lbuf[((size_t)(S_ - 2) * N_ + n) * C_ + tid];
    __syncthreads();

    for (int t = S_ - 2; t >= 0; --t) {
        float npv = np_nx, lv = l_nx;
        if (t > 0) {  // prefetch next (path-independent)
            np_nx = npot[((size_t)n * S_ + t - 1) * C_ + tid];
            l_nx  = lbuf[((size_t)(t - 1) * N_ + n) * C_ + tid];
        }
        float sc = npv + lv + Tt[pc * C_ + tid];  // + T[c][pc]
        sred[tid] = sc;
        __syncthreads();

        if (tid < 32) {
            float v = sred[tid];
            int   i = tid;
#pragma unroll
            for (int q = 1; q < 4; ++q) {         // increasing index: > keeps first
                float v2 = sred[tid + 32 * q];
                if (v2 > v) { v = v2; i = tid + 32 * q; }
            }
#pragma unroll
            for (int off = 16; off >= 1; off >>= 1) {
                float v2 = __shfl_down(v, off);
                int   i2 = __shfl_down(i, off);
                if (v2 > v || (v2 == v && i2 < i)) { v = v2; i = i2; }
            }
            if (tid == 0) pcsh = i;
        }
        __syncthreads();

        int pcn   = pcsh;
        float mt1 = mlds[t + 1];
        int pc2   = (mt1 != 0.0f) ? pcn : pc;
        if (tid == 0) {
            float mt = mlds[t];
            pred[n * S_ + t] = (mt != 0.0f) ? (float)pc2 : -1.0f;
        }
        pc = pc2;
    }
}

// ---------------------------------------------------------------------------
// Launch: inputs {X, pad_mask, W, T}; out = [prediction (N*S) | score (N)] f32.
// ws layout: npot (N*S*C f32) | l (S*N*C f32) | T^T (C*C f32) | prevc (N i32)
// ~64 MB total.
// ---------------------------------------------------------------------------
extern "C" void kernel_launch(void* const* d_in, const int* in_sizes, int n_in,
                              void* d_out, int out_size, void* d_ws, size_t ws_size,
                              hipStream_t stream) {
    (void)in_sizes; (void)n_in; (void)out_size; (void)ws_size;
    const float* X    = (const float*)d_in[0];
    const float* mask = (const float*)d_in[1];
    const float* W    = (const float*)d_in[2];
    const float* Tg   = (const float*)d_in[3];

    float* pred  = (float*)d_out;
    float* score = pred + (size_t)N_ * S_;

    float* npot  = (float*)d_ws;
    float* lbuf  = npot + (size_t)N_ * S_ * C_;
    float* Tt    = lbuf + (size_t)S_ * N_ * C_;
    int*   prevc = (int*)(Tt + C_ * C_);

    crf_transpose<<<C_, C_, 0, stream>>>(Tg, Tt);
    crf_gemm<<<(N_ * S_ / 16) / 8, 256, 0, stream>>>(X, W, npot);
    crf_forward<<<N_, 256, 0, stream>>>(npot, mask, Tg, lbuf, pred, score, prevc);
    crf_backtrace<<<N_, 128, 0, stream>>>(npot, lbuf, Tt, mask, prevc, pred);
}